// FPS_kNN_87084756893889
// MI455X (gfx1250) — compile-verified
//
#include <hip/hip_runtime.h>
#include <hip/hip_bf16.h>

// Problem constants (from reference setup_inputs)
#define B_ 8
#define N_ 8192
#define C_ 128
#define M_ 2048      // GROUP_NUM
#define K_ 32        // K_NEIGHBORS

typedef float v2f __attribute__((ext_vector_type(2)));
typedef float v8f __attribute__((ext_vector_type(8)));

// ---------------------------------------------------------------------------
// Kernel 1: Furthest Point Sampling. One workgroup per batch; point coords and
// running min-dist live in registers (8 points / thread); two-level LDS argmax
// per iteration. Sequential in m (2047 steps) by algorithmic necessity.
// ---------------------------------------------------------------------------
__global__ __launch_bounds__(1024) void fps_kernel(const float* __restrict__ xyz,
                                                   int* __restrict__ fps_idx) {
  const int b = blockIdx.x;
  const int t = threadIdx.x;
  const float* X = xyz + (size_t)b * N_ * 3;
  float px[8], py[8], pz[8], dst[8];
#pragma unroll
  for (int i = 0; i < 8; ++i) {
    const int n = i * 1024 + t;
    px[i] = X[n * 3 + 0];
    py[i] = X[n * 3 + 1];
    pz[i] = X[n * 3 + 2];
    dst[i] = 1e10f;
  }
  __shared__ float s_last[3];
  __shared__ float s_val[32];
  __shared__ int   s_idx[32];
  if (t == 0) {
    fps_idx[b * M_] = 0;                       // seeded at point 0
    s_last[0] = X[0]; s_last[1] = X[1]; s_last[2] = X[2];
  }
  __syncthreads();
  const int lane = t & 31, warp = t >> 5;
  for (int m = 1; m < M_; ++m) {
    const float lx = s_last[0], ly = s_last[1], lz = s_last[2];
    float bv = -1.0f; int bi = 0;
#pragma unroll
    for (int i = 0; i < 8; ++i) {
      const float dx = px[i] - lx, dy = py[i] - ly, dz = pz[i] - lz;
      const float d = dx * dx + dy * dy + dz * dz;
      dst[i] = fminf(dst[i], d);
      const int n = i * 1024 + t;
      if (dst[i] > bv || (dst[i] == bv && n < bi)) { bv = dst[i]; bi = n; }
    }
    // wave32 argmax (tie -> smallest index, matching jnp.argmax)
    for (int off = 16; off; off >>= 1) {
      const float ov = __shfl_xor(bv, off);
      const int   oi = __shfl_xor(bi, off);
      if (ov > bv || (ov == bv && oi < bi)) { bv = ov; bi = oi; }
    }
    if (lane == 0) { s_val[warp] = bv; s_idx[warp] = bi; }
    __syncthreads();
    if (t < 32) {
      bv = s_val[t]; bi = s_idx[t];
      for (int off = 16; off; off >>= 1) {
        const float ov = __shfl_xor(bv, off);
        const int   oi = __shfl_xor(bi, off);
        if (ov > bv || (ov == bv && oi < bi)) { bv = ov; bi = oi; }
      }
      if (t == 0) {
        fps_idx[b * M_ + m] = bi;
        s_last[0] = X[bi * 3 + 0];
        s_last[1] = X[bi * 3 + 1];
        s_last[2] = X[bi * 3 + 2];
      }
    }
    __syncthreads();
  }
}

// ---------------------------------------------------------------------------
// Kernel 2: build WMMA B-operand source P4[b][n] = (px, py, pz, |p|^2)
// ---------------------------------------------------------------------------
__global__ void prep_p4(const float* __restrict__ xyz, float4* __restrict__ P4) {
  const int i = blockIdx.x * blockDim.x + threadIdx.x;
  if (i >= B_ * N_) return;
  const float x = xyz[i * 3 + 0], y = xyz[i * 3 + 1], z = xyz[i * 3 + 2];
  P4[i] = make_float4(x, y, z, x * x + y * y + z * z);
}

// ---------------------------------------------------------------------------
// Kernel 3: gather lc_xyz / lc_x / lc_rgb and build WMMA A-operand source
// Q4[b][m] = (-2qx, -2qy, -2qz, 1).  sqdist row-offset |q|^2 is dropped: it
// does not change the per-row top-k ordering.
// ---------------------------------------------------------------------------
__global__ __launch_bounds__(128) void gather_lc(const float* __restrict__ xyz,
                                                 const float* __restrict__ x,
                                                 const float* __restrict__ rgb,
                                                 const int* __restrict__ fps_idx,
                                                 float* __restrict__ lc_xyz,
                                                 float* __restrict__ lc_x,
                                                 float* __restrict__ lc_rgb,
                                                 float4* __restrict__ Q4) {
  const int bm = blockIdx.x;           // b*M + m
  const int b = bm / M_;
  const int idx = fps_idx[bm];
  const int c = threadIdx.x;
  lc_x[(size_t)bm * C_ + c] = x[((size_t)b * N_ + idx) * C_ + c];
  if (c < 3) {
    lc_xyz[bm * 3 + c] = xyz[((size_t)b * N_ + idx) * 3 + c];
    lc_rgb[bm * 3 + c] = rgb[((size_t)b * N_ + idx) * 3 + c];
  }
  if (c == 0) {
    const size_t base = ((size_t)b * N_ + idx) * 3;
    const float qx = xyz[base], qy = xyz[base + 1], qz = xyz[base + 2];
    Q4[bm] = make_float4(-2.0f * qx, -2.0f * qy, -2.0f * qz, 1.0f);
  }
}

// ---------------------------------------------------------------------------
// Kernel 4: fused WMMA distance + wave-resident top-32 selection.
// Block = 16 queries, 512 threads = 16 waves.  Per 256-point chunk each wave
// computes one V_WMMA_F32_16X16X4_F32 tile (D = -2 q.p + |p|^2, exact fp32),
// spills C to LDS, then each wave scans its query row keeping the 32 current
// best in one register per lane (insertions serialized via ballot; they become
// rare as the running 32nd-best threshold tightens).
// ---------------------------------------------------------------------------
__global__ __launch_bounds__(512) void knn_kernel(const float4* __restrict__ P4,
                                                  const float4* __restrict__ Q4,
                                                  int* __restrict__ knn_idx) {
  const int blk = blockIdx.x;
  const int b = blk / (M_ / 16);
  const int qbase = (blk % (M_ / 16)) * 16;
  const int wave = threadIdx.x >> 5;
  const int lane = threadIdx.x & 31;
  const int half = lane >> 4;   // ISA 7.12.2: lanes 0-15 hold K={0,1}, 16-31 hold K={2,3}
  const int l16 = lane & 15;

  // A fragment (16x4 f32): row = lane%16 = query within tile
  const float4 q = Q4[b * M_ + qbase + l16];
  v2f afrag;
  afrag.x = half ? q.z : q.x;
  afrag.y = half ? q.w : q.y;

  __shared__ float sdist[16][256];   // 16 KB

  float best_d = 1e30f; int best_i = 0;   // this lane's entry of the top-32 set
  float cmax = 1e30f;  int cml = 0;       // current 32nd-best (wave-uniform)

  for (int chunk = 0; chunk < N_; chunk += 256) {
    // B fragment (4x16 f32): col = lane%16, this wave's 16-column slice
    const float4 p = P4[(size_t)b * N_ + chunk + wave * 16 + l16];
    v2f bfrag;
    bfrag.x = half ? p.z : p.x;
    bfrag.y = half ? p.w : p.y;
    v8f acc = {};
    acc = __builtin_amdgcn_wmma_f32_16x16x4_f32(
        /*neg_a=*/false, afrag, /*neg_b=*/false, bfrag,
        /*c_mod=*/(short)0, acc, /*reuse_a=*/false, /*reuse_b=*/false);
    // C layout (ISA 7.12.2): VGPR v, lanes 0-15 -> row v; lanes 16-31 -> row v+8
    const int coff = wave * 16 + l16;
    const int rbase = half * 8;
#pragma unroll
    for (int v = 0; v < 8; ++v) sdist[rbase + v][coff] = acc[v];
    __syncthreads();

    // selection: wave w owns query row w; 256 candidates in 8 groups of 32
#pragma unroll
    for (int g = 0; g < 8; ++g) {
      const int ci = g * 32 + lane;
      const float d = sdist[wave][ci];
      const int pid = chunk + ci;
      unsigned mask = (unsigned)__ballot(d < cmax);
      while (mask) {
        const int src = __ffs(mask) - 1;
        mask &= mask - 1;
        const float dj = __shfl(d, src);
        const int   ij = __shfl(pid, src);
        if (dj < cmax) {                        // wave-uniform branch
          if (lane == cml) { best_d = dj; best_i = ij; }
          // recompute wave max of the 32-set (value + holder lane)
          float v = best_d; int l = lane;
          for (int off = 16; off; off >>= 1) {
            const float ov = __shfl_xor(v, off);
            const int   ol = __shfl_xor(l, off);
            if (ov > v || (ov == v && ol < l)) { v = ov; l = ol; }
          }
          cmax = v; cml = l;
        }
      }
    }
    __syncthreads();
  }

  // bitonic sort the 32 (d,i) pairs ascending across lanes (matches top_k order)
  for (int k = 2; k <= 32; k <<= 1) {
    for (int j = k >> 1; j >= 1; j >>= 1) {
      const float od = __shfl_xor(best_d, j);
      const int   oi = __shfl_xor(best_i, j);
      const bool up = ((lane & k) == 0);
      const bool lower = ((lane & j) == 0);
      const bool takeMin = (lower == up);
      const bool otherSmaller = (od < best_d) || (od == best_d && oi < best_i);
      if (takeMin ? otherSmaller : !otherSmaller) { best_d = od; best_i = oi; }
    }
  }
  knn_idx[((size_t)(b * M_ + qbase + wave)) * K_ + lane] = best_i;
}

// ---------------------------------------------------------------------------
// Kernel 5: gather knn_xyz / knn_x / knn_rgb (pure bandwidth, coalesced rows)
// ---------------------------------------------------------------------------
__global__ __launch_bounds__(128) void gather_knn(const float* __restrict__ xyz,
                                                  const float* __restrict__ x,
                                                  const float* __restrict__ rgb,
                                                  const int* __restrict__ knn_idx,
                                                  float* __restrict__ knn_xyz,
                                                  float* __restrict__ knn_x,
                                                  float* __restrict__ knn_rgb) {
  const int bm = blockIdx.x;
  const int b = bm / M_;
  const int c = threadIdx.x;
  for (int k = 0; k < K_; ++k) {
    const size_t row = (size_t)bm * K_ + k;
    const int idx = knn_idx[row];
    knn_x[row * C_ + c] = x[((size_t)b * N_ + idx) * C_ + c];
    if (c < 3) {
      knn_xyz[row * 3 + c] = xyz[((size_t)b * N_ + idx) * 3 + c];
      knn_rgb[row * 3 + c] = rgb[((size_t)b * N_ + idx) * 3 + c];
    }
  }
}

// ---------------------------------------------------------------------------
extern "C" void kernel_launch(void* const* d_in, const int* in_sizes, int n_in,
                              void* d_out, int out_size, void* d_ws, size_t ws_size,
                              hipStream_t stream) {
  const float* xyz = (const float*)d_in[0];
  const float* x   = (const float*)d_in[1];
  const float* rgb = (const float*)d_in[2];

  // workspace carve-out (~3.4 MB total, 16B-aligned segments)
  char* ws = (char*)d_ws;
  float4* P4 = (float4*)(ws);                                         // B*N*16  = 1 MB
  float4* Q4 = (float4*)(ws + (size_t)B_ * N_ * 16);                  // B*M*16  = 256 KB
  int* fps_idx = (int*)(ws + (size_t)B_ * N_ * 16 + (size_t)B_ * M_ * 16);
  int* knn_idx = fps_idx + B_ * M_;                                   // B*M*K*4 = 2 MB

  // output layout: tuple concatenated flat, all float32
  float* out = (float*)d_out;
  float* lc_xyz  = out;
  float* lc_x    = lc_xyz  + (size_t)B_ * M_ * 3;
  float* lc_rgb  = lc_x    + (size_t)B_ * M_ * C_;
  float* knn_xyz = lc_rgb  + (size_t)B_ * M_ * 3;
  float* knn_x   = knn_xyz + (size_t)B_ * M_ * K_ * 3;
  float* knn_rgb = knn_x   + (size_t)B_ * M_ * K_ * C_;

  fps_kernel<<<B_, 1024, 0, stream>>>(xyz, fps_idx);
  prep_p4<<<(B_ * N_ + 255) / 256, 256, 0, stream>>>(xyz, P4);
  gather_lc<<<B_ * M_, 128, 0, stream>>>(xyz, x, rgb, fps_idx, lc_xyz, lc_x, lc_rgb, Q4);
  knn_kernel<<<B_ * (M_ / 16), 512, 0, stream>>>(P4, Q4, knn_idx);
  gather_knn<<<B_ * M_, 128, 0, stream>>>(xyz, x, rgb, knn_idx, knn_xyz, knn_x, knn_rgb);
}